// WaveletNoise_12051678233393
// MI455X (gfx1250) — compile-verified
//
#include <hip/hip_runtime.h>
#include <hip/hip_bf16.h>
#include <math.h>

typedef __attribute__((ext_vector_type(2))) float v2f;
typedef __attribute__((ext_vector_type(8))) float v8f;

// grid coordinate: linspace(-1, 1, S)[i]  (S==1 -> -1), division folded to fma
template <int S>
__device__ __forceinline__ float gridcT(int i) {
    if constexpr (S == 1) {
        return -1.0f;
    } else {
        constexpr float step = 2.0f / (float)(S - 1);
        return fmaf((float)i, step, -1.0f);
    }
}

// Separable level via f32 WMMA, K-chunk outer so each y-basis chunk is computed once:
//   out[b] += sum_i ux[b,i]*(sum_j c[i,j] ey[b,j]) + sum_i ex[b,i]*(sum_j c[i,j] vy[b,j])  (RICKER)
//   out[b] += sum_i ex[b,i]*(sum_j c[i,j] ey[b,j])                                          (Gaussian)
// cT2 in LDS holds c transposed with K-pairs interleaved (8-byte B operands).
// One wave handles 16 points (M=16); NT = S/16 N-tiles kept live in register accumulators.
// Each lane accumulates its N-column's contribution to point (r + 8*h) into part[r];
// one batch of LDS float atomics at kernel end merges the 16 N-lanes per point.
template <int S, bool RICKER>
__device__ __forceinline__ void level_sep(float py, float inv_s2,
                                          const v2f* __restrict__ cT2,
                                          const float* __restrict__ pxp,
                                          float* __restrict__ part, int lane) {
    constexpr int NT = S / 16;
    const int m = lane & 15;   // N-lane within tile
    const int h = lane >> 4;   // half-wave

    const v8f vzero = {0.f, 0.f, 0.f, 0.f, 0.f, 0.f, 0.f, 0.f};
    v8f c1[NT];
    v8f c2[NT];
#pragma unroll
    for (int t = 0; t < NT; ++t) {
        c1[t] = vzero;
        if (RICKER) c2[t] = vzero;
    }

#pragma unroll
    for (int k0 = 0; k0 < S; k0 += 4) {
        // A-matrix (f32 16x4): lanes 0-15 hold K=k0,k0+1; lanes 16-31 K=k0+2,k0+3
        const int j0 = k0 + 2 * h;
        float d0 = py - gridcT<S>(j0);
        float d1 = py - gridcT<S>(j0 + 1);
        float r0 = d0 * d0 * inv_s2;
        float r1 = d1 * d1 * inv_s2;
        float e0 = __expf(-0.5f * r0);
        float e1 = __expf(-0.5f * r1);

        v2f aE; aE.x = e0; aE.y = e1;
        v2f aV;
        if (RICKER) { aV.x = (1.0f - r0) * e0; aV.y = (1.0f - r1) * e1; }

#pragma unroll
        for (int t = 0; t < NT; ++t) {
            // B-matrix (4x16): pair (j0, j0+1) is one 8-byte LDS word
            v2f b = cT2[(k0 / 2 + h) * S + t * 16 + m];
            c1[t] = __builtin_amdgcn_wmma_f32_16x16x4_f32(false, aE, false, b,
                                                          (short)0, c1[t], false, false);
            if (RICKER)
                c2[t] = __builtin_amdgcn_wmma_f32_16x16x4_f32(false, aV, false, b,
                                                              (short)0, c2[t], false, false);
        }
    }

    // Combine D with x-basis: D VGPR r of tile t holds point (r + 8*h), column t*16+m.
#pragma unroll
    for (int t = 0; t < NT; ++t) {
        const float cxv = gridcT<S>(t * 16 + m);
#pragma unroll
        for (int r = 0; r < 8; ++r) {
            float d = pxp[r] - cxv;
            float rx = d * d * inv_s2;
            float ex = __expf(-0.5f * rx);
            if (RICKER) part[r] += ex * fmaf(1.0f - rx, c1[t][r], c2[t][r]);
            else        part[r] += ex * c1[t][r];
        }
    }
}

__global__ __launch_bounds__(256) void
WaveletNoise_12051678233393_kernel(const float* __restrict__ coords,
                                   const float* __restrict__ scal,
                                   const float* __restrict__ det0,
                                   const float* __restrict__ det1,
                                   const float* __restrict__ det2,
                                   const float* __restrict__ det3,
                                   const float* __restrict__ det4,
                                   const float* __restrict__ det5,
                                   const float* __restrict__ det6,
                                   float* __restrict__ out) {
    // Pair-interleaved transposed coefficient tiles:
    // scaling @0 (16x16), L4 @256 (16x16), L5 @512 (32x32), L6 @1536 (64x64)
    __shared__ float  sCT[5632];
    __shared__ float4 sTab[85];        // flattened small-level centers {cx, cy, inv_s2, coeff}
    __shared__ float  sAcc[8][16];

    const int tid  = threadIdx.x;
    const int lane = tid & 31;
    const int wave = tid >> 5;
    const int m    = lane & 15;
    const int h    = lane >> 4;

    // Stage with transpose + K-pair interleave: dst[((j>>1)*S + i)*2 + (j&1)] = c[i*S + j]
    {
        int idx = tid;                     // 256 elements, one per thread
        int j = idx >> 4, i = idx & 15;
        int d16 = ((j >> 1) * 16 + i) * 2 + (j & 1);
        sCT[d16]       = scal[i * 16 + j];
        sCT[256 + d16] = det4[i * 16 + j];
    }
    for (int idx = tid; idx < 1024; idx += 256) {
        int j = idx >> 5, i = idx & 31;
        sCT[512 + ((j >> 1) * 32 + i) * 2 + (j & 1)] = det5[i * 32 + j];
    }
    for (int idx = tid; idx < 4096; idx += 256) {
        int j = idx >> 6, i = idx & 63;
        sCT[1536 + ((j >> 1) * 64 + i) * 2 + (j & 1)] = det6[i * 64 + j];
    }
    // Flattened table for direct levels 0-3 (1 + 4 + 16 + 64 = 85 centers)
    if (tid < 85) {
        int L, base;
        if (tid == 0)       { L = 0; base = 0;  }
        else if (tid < 5)   { L = 1; base = 1;  }
        else if (tid < 21)  { L = 2; base = 5;  }
        else                { L = 3; base = 21; }
        int S = 1 << L;
        int o = tid - base;
        int i = o >> L, j = o & (S - 1);
        float sig  = 0.9f / (float)S;
        float inv  = 1.0f / (sig * sig + 1e-8f);
        float step = (S == 1) ? 0.0f : 2.0f / (float)(S - 1);
        float cx   = (S == 1) ? -1.0f : fmaf((float)i, step, -1.0f);
        float cy   = (S == 1) ? -1.0f : fmaf((float)j, step, -1.0f);
        const float* cp = (L == 0) ? det0 : (L == 1) ? det1 : (L == 2) ? det2 : det3;
        sTab[tid] = make_float4(cx, cy, inv, cp[o]);
    }
    if (lane < 16) sAcc[wave][lane] = 0.0f;
    __syncthreads();

    const int p = blockIdx.x * 128 + wave * 16 + m;   // 8 waves * 16 pts per block
    const float px = coords[2 * p]     * 0.1f;        // xy = coords / DOMAIN
    const float py = coords[2 * p + 1] * 0.1f;

    // Broadcast x coords of the 8 points this half-wave's D rows cover (level-invariant):
    // point (r + 8*h) lives in lane (r + 8*h); raw bpermute, no clamp code.
    float pxp[8];
#pragma unroll
    for (int r = 0; r < 8; ++r) {
        int src = (r + 8 * h) << 2;   // byte index
        pxp[r] = __int_as_float(__builtin_amdgcn_ds_bpermute(src, __float_as_int(px)));
    }

    float* acc = sAcc[wave];

    // Direct small levels: the two lanes sharing point m split the 85 centers even/odd
    // (uniform control flow, no half-wave serialization).
    {
        float dsum = 0.0f;
        for (int t = 0; t < 43; ++t) {
            int idx = 2 * t + h;
            if (idx < 85) {
                float4 e = sTab[idx];
                float dx = px - e.x;
                float dy = py - e.y;
                float r2 = (dx * dx + dy * dy) * e.z;
                dsum += e.w * (2.0f - r2) * __expf(-0.5f * r2);
            }
        }
        atomicAdd(&acc[m], dsum);
    }

    const v2f* cT2 = (const v2f*)sCT;

    // Per-lane partials for points (r + 8*h), accumulated across all WMMA levels
    float part[8] = {0.f, 0.f, 0.f, 0.f, 0.f, 0.f, 0.f, 0.f};

    // scaling: sigma2 = (0.9*1.6)^2 = 2.0736 (no epsilon), plain Gaussian
    level_sep<16, false>(py, 1.0f / (1.44f * 1.44f),                   cT2,       pxp, part, lane);
    // level 4: sigma = 0.9/16
    level_sep<16, true >(py, 1.0f / (0.05625f   * 0.05625f   + 1e-8f), cT2 + 128, pxp, part, lane);
    // level 5: sigma = 0.9/32
    level_sep<32, true >(py, 1.0f / (0.028125f  * 0.028125f  + 1e-8f), cT2 + 256, pxp, part, lane);
    // level 6: sigma = 0.9/64
    level_sep<64, true >(py, 1.0f / (0.0140625f * 0.0140625f + 1e-8f), cT2 + 768, pxp, part, lane);

    // Merge the 16 N-lanes per point: 8 float LDS atomics per lane, once per kernel
    {
        float* accH = acc + 8 * h;
#pragma unroll
        for (int r = 0; r < 8; ++r) atomicAdd(&accH[r], part[r]);
    }

    __syncthreads();

    // softplus + store (lanes 0-15 own the 16 points)
    if (h == 0) {
        float x = acc[m];
        out[p] = (x > 20.0f) ? x : log1pf(__expf(x));
    }
}

extern "C" void kernel_launch(void* const* d_in, const int* in_sizes, int n_in,
                              void* d_out, int out_size, void* d_ws, size_t ws_size,
                              hipStream_t stream) {
    const float* coords = (const float*)d_in[0];
    const float* scal   = (const float*)d_in[1];
    const float* det0   = (const float*)d_in[2];
    const float* det1   = (const float*)d_in[3];
    const float* det2   = (const float*)d_in[4];
    const float* det3   = (const float*)d_in[5];
    const float* det4   = (const float*)d_in[6];
    const float* det5   = (const float*)d_in[7];
    const float* det6   = (const float*)d_in[8];
    float* out = (float*)d_out;

    const int n_points = in_sizes[0] / 2;        // 32768
    const int blocks = n_points / 128;           // 8 waves * 16 points per block
    WaveletNoise_12051678233393_kernel<<<blocks, 256, 0, stream>>>(
        coords, scal, det0, det1, det2, det3, det4, det5, det6, out);
}